// GraphTransformerBlock2_75557064671478
// MI455X (gfx1250) — compile-verified
//
#include <hip/hip_runtime.h>

#define HH 3
#define CC 64
#define HC 192
#define EDIM 5
#define NEG_SLOPE 0.2f
#define LN_EPS 1e-5f

typedef float v2f __attribute__((ext_vector_type(2)));
typedef float v8f __attribute__((ext_vector_type(8)));

// ---- monotonic float<->uint encoding so atomicMax(uint) == float max ----
__device__ __forceinline__ unsigned f2mono(float f) {
  unsigned u = __float_as_uint(f);
  return (u & 0x80000000u) ? ~u : (u | 0x80000000u);
}
__device__ __forceinline__ float mono2f(unsigned u) {
  return (u & 0x80000000u) ? __uint_as_float(u & 0x7FFFFFFFu)
                           : __uint_as_float(~u);
}

__global__ void zero_f32(float* __restrict__ p, long long n) {
  long long i = (long long)blockIdx.x * blockDim.x + threadIdx.x;
  if (i < n) p[i] = 0.0f;
}

// w_e_att[d][h] = sum_c lin_edge_w[d, h*64+c] * att_edge[h, c]   (5x3)
__global__ void weatt_kernel(const float* __restrict__ lin_edge_w,
                             const float* __restrict__ att_edge,
                             float* __restrict__ weatt) {
  int i = threadIdx.x;
  if (i >= EDIM * HH) return;
  int d = i / HH, h = i % HH;
  float s = 0.0f;
  for (int c = 0; c < CC; ++c)
    s += lin_edge_w[d * HC + h * CC + c] * att_edge[h * CC + c];
  weatt[i] = s;
}

// segment-sum of edge_attr over dst + counts (for self-loop fill 'mean')
__global__ void edge_attr_sum(const float* __restrict__ ea,
                              const int* __restrict__ ei,
                              float* __restrict__ attrsum,
                              float* __restrict__ cnt, int E) {
  int e = blockIdx.x * blockDim.x + threadIdx.x;
  if (e >= E) return;
  int d = ei[E + e];
  for (int k = 0; k < EDIM; ++k) atomicAdd(&attrsum[d * EDIM + k], ea[e * EDIM + k]);
  atomicAdd(&cnt[d], 1.0f);
}

__global__ void loop_attr_div(float* __restrict__ attrsum,
                              const float* __restrict__ cnt, int N) {
  int i = blockIdx.x * blockDim.x + threadIdx.x;
  if (i >= N * EDIM) return;
  attrsum[i] /= fmaxf(cnt[i / EDIM], 1.0f);
}

// ---- f32 WMMA GEMM with LDS-staged B: Cout[M,NSZ] = (A + Abias) @ B + Obias
// Block = 256 threads = 8 waves. The whole B matrix (KSZ x NSZ) plus the
// optional A-bias is staged once into LDS with coalesced loads; each wave
// then computes one 16-row M tile across all NSZ/16 N tiles, feeding
// V_WMMA_F32_16X16X4_F32 from ds_load (B) + global_load_b64 (A).
// A-layout (16x4 f32): lanes 0-15 carry M rows w/ K={0,1}; lanes 16-31 K={2,3}.
// B-layout (4x16 f32): V0 = K=0 row (lanes 0-15) / K=2 (lanes 16-31), V1 = K=1/K=3.
// D-layout: VGPR j -> (M=j, N=lane) lanes 0-15 ; (M=j+8, N=lane-16) lanes 16-31.
template <int KSZ, int NSZ, bool ABIAS, bool OBIAS>
__global__ void wmma_gemm_lds(const float* __restrict__ A,
                              const float* __restrict__ Abias,
                              const float* __restrict__ B,
                              const float* __restrict__ Obias,
                              float* __restrict__ Cout, int M) {
  __shared__ float sB[KSZ * NSZ + (ABIAS ? KSZ : 1)];
  float* sAb = sB + KSZ * NSZ;

  // cooperative staging of B (+ Abias) into LDS, coalesced
  for (int i = threadIdx.x; i < KSZ * NSZ; i += 256) sB[i] = B[i];
  if (ABIAS)
    for (int i = threadIdx.x; i < KSZ; i += 256) sAb[i] = Abias[i];
  __syncthreads();

  const int wave = threadIdx.x >> 5;     // 0..7
  const int lane = threadIdx.x & 31;     // full wave, EXEC all-ones at WMMA
  const int half = lane >> 4;
  const int l15  = lane & 15;
  const int mtile = blockIdx.x * 8 + wave;
  if (mtile * 16 >= M) return;           // wave-uniform exit

  int rowA = mtile * 16 + l15;
  if (rowA >= M) rowA = M - 1;           // clamp (M % 16 == 0 here, no-op)
  const float* Arow = A + (size_t)rowA * KSZ;

  for (int nt = 0; nt < NSZ / 16; ++nt) {
    const int col = nt * 16 + l15;
    v8f c = {};
    for (int k = 0; k < KSZ; k += 4) {
      const int ka = k + 2 * half;
      float a0 = Arow[ka];
      float a1 = Arow[ka + 1];
      if (ABIAS) { a0 += sAb[ka]; a1 += sAb[ka + 1]; }
      v2f a; a.x = a0; a.y = a1;
      v2f b; b.x = sB[ka * NSZ + col];
             b.y = sB[(ka + 1) * NSZ + col];
      c = __builtin_amdgcn_wmma_f32_16x16x4_f32(false, a, false, b,
                                                (short)0, c, false, false);
    }
    const int rowBase = mtile * 16 + half * 8;
    const float ob = OBIAS ? Obias[col] : 0.0f;
#pragma unroll
    for (int j = 0; j < 8; ++j) {
      const int r = rowBase + j;
      if (r < M) Cout[(size_t)r * NSZ + col] = c[j] + ob;
    }
  }
}

// a_src[n,h], a_dst[n,h] from xs
__global__ void attn_node(const float* __restrict__ xs,
                          const float* __restrict__ att_src,
                          const float* __restrict__ att_dst,
                          float* __restrict__ asrc, float* __restrict__ adst,
                          int N) {
  int i = blockIdx.x * blockDim.x + threadIdx.x;
  if (i >= N * HH) return;
  int n = i / HH, h = i % HH;
  const float* xr = xs + (size_t)n * HC + h * CC;
  float s1 = 0.0f, s2 = 0.0f;
  for (int cix = 0; cix < CC; ++cix) {
    float v = xr[cix];
    s1 += v * att_src[h * CC + cix];
    s2 += v * att_dst[h * CC + cix];
  }
  asrc[i] = s1;
  adst[i] = s2;
}

__global__ void edge_logits(const float* __restrict__ ea,
                            const int* __restrict__ ei,
                            const float* __restrict__ asrc,
                            const float* __restrict__ adst,
                            const float* __restrict__ weatt,
                            float* __restrict__ exbuf,
                            unsigned* __restrict__ lmax, int E) {
  int e = blockIdx.x * blockDim.x + threadIdx.x;
  if (e >= E) return;
  int s = ei[e], d = ei[E + e];
  float at[EDIM];
  for (int k = 0; k < EDIM; ++k) at[k] = ea[(size_t)e * EDIM + k];
  for (int h = 0; h < HH; ++h) {
    float l = asrc[s * HH + h] + adst[d * HH + h];
    for (int k = 0; k < EDIM; ++k) l += at[k] * weatt[k * HH + h];
    l = (l >= 0.0f) ? l : NEG_SLOPE * l;
    exbuf[(size_t)e * HH + h] = l;
    atomicMax(&lmax[d * HH + h], f2mono(l));
  }
}

__global__ void loop_logits(const float* __restrict__ loopattr,
                            const float* __restrict__ asrc,
                            const float* __restrict__ adst,
                            const float* __restrict__ weatt,
                            float* __restrict__ exbuf,
                            unsigned* __restrict__ lmax, int N, int E) {
  int n = blockIdx.x * blockDim.x + threadIdx.x;
  if (n >= N) return;
  float at[EDIM];
  for (int k = 0; k < EDIM; ++k) at[k] = loopattr[(size_t)n * EDIM + k];
  for (int h = 0; h < HH; ++h) {
    float l = asrc[n * HH + h] + adst[n * HH + h];
    for (int k = 0; k < EDIM; ++k) l += at[k] * weatt[k * HH + h];
    l = (l >= 0.0f) ? l : NEG_SLOPE * l;
    exbuf[(size_t)(E + n) * HH + h] = l;
    atomicMax(&lmax[n * HH + h], f2mono(l));
  }
}

__global__ void edge_exp(const int* __restrict__ ei,
                         float* __restrict__ exbuf,
                         const unsigned* __restrict__ lmax,
                         float* __restrict__ denom, int E, int N) {
  int i = blockIdx.x * blockDim.x + threadIdx.x;
  int tot = (E + N) * HH;
  if (i >= tot) return;
  int idx = i / HH, h = i % HH;
  int d = (idx < E) ? ei[E + idx] : (idx - E);
  float ex = expf(exbuf[i] - mono2f(lmax[d * HH + h]));
  exbuf[i] = ex;
  atomicAdd(&denom[d * HH + h], ex);
}

// one 192-thread block per edge: out[dst] += alpha * xs[src]
__global__ void aggregate(const int* __restrict__ ei,
                          const float* __restrict__ exbuf,
                          const float* __restrict__ denom,
                          const float* __restrict__ xs,
                          float* __restrict__ outacc, int E, int N) {
  int idx = blockIdx.x;
  int t = threadIdx.x;               // 0..191
  int s, d;
  if (idx < E) { s = ei[idx]; d = ei[E + idx]; }
  else         { s = d = idx - E; }
  int h = t >> 6;
  float alpha = exbuf[(size_t)idx * HH + h] / denom[d * HH + h];
  atomicAdd(&outacc[(size_t)d * HC + t], alpha * xs[(size_t)s * HC + t]);
}

// out = LayerNorm(a + b) * g + beta, row length 64, one wave per row
__global__ void ln_residual(const float* __restrict__ a,
                            const float* __restrict__ b,
                            const float* __restrict__ g,
                            const float* __restrict__ beta,
                            float* __restrict__ out) {
  int n = blockIdx.x;
  int t = threadIdx.x;               // 0..31
  size_t base = (size_t)n * CC;
  float v0 = a[base + t]      + b[base + t];
  float v1 = a[base + t + 32] + b[base + t + 32];
  float s = v0 + v1;
  for (int off = 16; off > 0; off >>= 1) s += __shfl_xor(s, off, 32);
  float m = s * (1.0f / CC);
  float d0 = v0 - m, d1 = v1 - m;
  float vs = d0 * d0 + d1 * d1;
  for (int off = 16; off > 0; off >>= 1) vs += __shfl_xor(vs, off, 32);
  float inv = rsqrtf(vs * (1.0f / CC) + LN_EPS);
  out[base + t]      = d0 * inv * g[t]      + beta[t];
  out[base + t + 32] = d1 * inv * g[t + 32] + beta[t + 32];
}

extern "C" void kernel_launch(void* const* d_in, const int* in_sizes, int n_in,
                              void* d_out, int out_size, void* d_ws, size_t ws_size,
                              hipStream_t stream) {
  const float* x          = (const float*)d_in[0];
  const float* edge_attr  = (const float*)d_in[1];
  const float* lin_w      = (const float*)d_in[2];
  const float* att_src    = (const float*)d_in[3];
  const float* att_dst    = (const float*)d_in[4];
  const float* lin_edge_w = (const float*)d_in[5];
  const float* att_edge   = (const float*)d_in[6];
  const float* gat_bias   = (const float*)d_in[7];
  const float* w1         = (const float*)d_in[8];
  const float* b1         = (const float*)d_in[9];
  const float* ln1_g      = (const float*)d_in[10];
  const float* ln1_b      = (const float*)d_in[11];
  const float* w2         = (const float*)d_in[12];
  const float* b2         = (const float*)d_in[13];
  const float* ln2_g      = (const float*)d_in[14];
  const float* ln2_b      = (const float*)d_in[15];
  const int*   ei         = (const int*)d_in[16];
  float* out = (float*)d_out;

  const int N  = in_sizes[0] / 64;      // 50000
  const int E  = in_sizes[1] / EDIM;    // 800000
  const int E2 = E + N;

  // workspace layout (floats)
  float* ws = (float*)d_ws;
  size_t off = 0;
  float* xs       = ws + off; off += (size_t)N * HC;       // x @ lin_w
  float* gat      = ws + off; off += (size_t)N * HC;       // GAT accumulator
  float* asrc     = ws + off; off += (size_t)N * HH;
  float* adst     = ws + off; off += (size_t)N * HH;
  unsigned* lmax  = (unsigned*)(ws + off); off += (size_t)N * HH;
  float* denom    = ws + off; off += (size_t)N * HH;
  float* attrsum  = ws + off; off += (size_t)N * EDIM;     // becomes loop_attr
  float* cnt      = ws + off; off += (size_t)N;
  float* weatt    = ws + off; off += 32;
  float* exbuf    = ws + off; off += (size_t)E2 * HH;
  float* y1       = ws + off; off += (size_t)N * CC;
  float* h1       = ws + off; off += (size_t)N * CC;
  float* y2       = ws + off; off += (size_t)N * CC;
  (void)ws_size; (void)n_in; (void)out_size;

  // zero contiguous region gat..cnt (covers lmax init: encoded 0 < any finite)
  long long nz = (long long)N * (HC + 4 * HH + EDIM + 1);
  zero_f32<<<(unsigned)((nz + 255) / 256), 256, 0, stream>>>(gat, nz);

  weatt_kernel<<<1, 32, 0, stream>>>(lin_edge_w, att_edge, weatt);
  edge_attr_sum<<<(E + 255) / 256, 256, 0, stream>>>(edge_attr, ei, attrsum, cnt, E);
  loop_attr_div<<<((N * EDIM) + 255) / 256, 256, 0, stream>>>(attrsum, cnt, N);

  const int mtiles = (N + 15) / 16;            // 3125
  const int gblocks = (mtiles + 7) / 8;        // 391

  // xs = x @ lin_w   (K=64, out width 192)
  wmma_gemm_lds<64, HC, false, false>
      <<<gblocks, 256, 0, stream>>>(x, nullptr, lin_w, nullptr, xs, N);

  attn_node<<<((N * HH) + 255) / 256, 256, 0, stream>>>(xs, att_src, att_dst, asrc, adst, N);
  edge_logits<<<(E + 255) / 256, 256, 0, stream>>>(edge_attr, ei, asrc, adst, weatt, exbuf, lmax, E);
  loop_logits<<<(N + 255) / 256, 256, 0, stream>>>(attrsum, asrc, adst, weatt, exbuf, lmax, N, E);
  edge_exp<<<((E2 * HH) + 255) / 256, 256, 0, stream>>>(ei, exbuf, lmax, denom, E, N);
  aggregate<<<E2, HC, 0, stream>>>(ei, exbuf, denom, xs, gat, E, N);

  // y1 = (gat + gat_bias) @ w1 + b1   (K=192, out width 64)
  wmma_gemm_lds<HC, CC, true, true>
      <<<gblocks, 256, 0, stream>>>(gat, gat_bias, w1, b1, y1, N);
  ln_residual<<<N, 32, 0, stream>>>(x, y1, ln1_g, ln1_b, h1);
  // y2 = h1 @ w2 + b2   (K=64, out width 64)
  wmma_gemm_lds<CC, CC, false, true>
      <<<gblocks, 256, 0, stream>>>(h1, nullptr, w2, b2, y2, N);
  ln_residual<<<N, 32, 0, stream>>>(y2, h1, ln2_g, ln2_b, out);
}